// TopKSAE_60292750901714
// MI455X (gfx1250) — compile-verified
//
#include <hip/hip_runtime.h>
#include <hip/hip_bf16.h>

#define H     2048
#define DSAE  16384
#define NROWS 8192
#define KTOP  163

#define BM 64
#define BN 256
#define KB 16

typedef __attribute__((ext_vector_type(2))) float v2f;
typedef __attribute__((ext_vector_type(8))) float v8f;
typedef __attribute__((ext_vector_type(4))) int   v4i;

#if defined(__has_builtin)
#  if __has_builtin(__builtin_amdgcn_global_load_async_to_lds_b128) && \
      __has_builtin(__builtin_amdgcn_s_wait_asynccnt)
#    define SAE_ASYNC 1
#  endif
#endif

// 16-byte global -> LDS copy: async-to-LDS on CDNA5 toolchains that expose the
// builtin, plain float4 staging otherwise. Both are used inside the same
// double-buffered pipeline (barrier at chunk start makes the overwrite safe).
__device__ __forceinline__ void cp16(const float* __restrict__ g, float* l) {
#ifdef SAE_ASYNC
    __builtin_amdgcn_global_load_async_to_lds_b128(
        (__attribute__((address_space(1))) v4i*)g,
        (__attribute__((address_space(3))) v4i*)l, 0, 0);
#else
    *(float4*)l = *(const float4*)g;
#endif
}

__device__ __forceinline__ void cp_wait() {
#ifdef SAE_ASYNC
    __builtin_amdgcn_s_wait_asynccnt(0);
#endif
}

// ---------------------------------------------------------------------------
// Kernel 1: transpose W_dec (H x DSAE) -> WdT (DSAE x H) so decoder gathers
// are contiguous per active latent (8KB rows, L2-resident: 128MB < 192MB L2).
// ---------------------------------------------------------------------------
__global__ __launch_bounds__(256) void sae_transpose_wdec(
    const float* __restrict__ Wd, float* __restrict__ WdT)
{
    __shared__ float tile[32][33];
    const int j0 = blockIdx.x * 32;   // DSAE dim
    const int h0 = blockIdx.y * 32;   // H dim
    const int t  = threadIdx.x;
    const int c  = t & 31;
    const int r0 = t >> 5;            // 0..7
#pragma unroll
    for (int i = 0; i < 4; ++i) {
        const int hh = r0 + i * 8;
        tile[hh][c] = Wd[(size_t)(h0 + hh) * DSAE + (j0 + c)];
    }
    __syncthreads();
#pragma unroll
    for (int i = 0; i < 4; ++i) {
        const int jj = r0 + i * 8;
        WdT[(size_t)(j0 + jj) * H + (h0 + c)] = tile[c][jj];
    }
}

// ---------------------------------------------------------------------------
// Kernel 2: encoder GEMM  latent = relu(x @ W_enc^T + b_enc)
// f32 WMMA 16x16x4. Block tile 64 rows x 256 latents, 8 waves (2x4 grid),
// each wave owns a 32x64 slab = eight 16x16 f32 accumulators (A-frag reused
// 4x, B-frag reused 2x -> 12 LDS dwords per 8 WMMAs). Double-buffered LDS
// with (optionally async) global->LDS staging.
// ---------------------------------------------------------------------------
__global__ __launch_bounds__(256) void sae_encode(
    const float* __restrict__ x, const float* __restrict__ W_enc,
    const float* __restrict__ b_enc, float* __restrict__ latent)
{
    __shared__ float As[2][BM][KB + 1];   //  8.5 KB
    __shared__ float Bs[2][BN][KB + 1];   // 34.0 KB

    const int t    = threadIdx.x;
    const int rowA = blockIdx.y * BM;
    const int colB = blockIdx.x * BN;
    const int w    = t >> 5;
    const int lane = t & 31;
    const int mw   = w >> 2;              // 0..1 : 32-row half
    const int nw   = w & 3;               // 0..3 : 64-col quarter

    v8f acc[2][4] = {};

    // staging pattern: 4 consecutive threads cover one 16-float row chunk
    const int ar = t >> 2;                // 0..63
    const int ac = (t & 3) << 2;          // 0,4,8,12
    const float* gA = x     + (size_t)(rowA + ar) * H + ac;
    const float* gB = W_enc + (size_t)(colB + ar) * H + ac;

    // prologue: chunk 0 -> buffer 0
    cp16(gA, &As[0][ar][ac]);
#pragma unroll
    for (int i = 0; i < 4; ++i)
        cp16(gB + (size_t)i * 64 * H, &Bs[0][ar + i * 64][ac]);

    const int m0 = mw * 32 + (lane & 15);       // A rows (and +16)
    const int nb = nw * 64 + (lane & 15);       // B rows (+j*16)
    const int kb = (lane >> 4) << 1;            // K sub-offset {0,2}

    const int NCHUNK = H / KB;                  // 128
    for (int c = 0; c < NCHUNK; ++c) {
        const int buf = c & 1;
        cp_wait();
        __syncthreads();          // chunk c resident; all reads of buf^1 done
        if (c + 1 < NCHUNK) {
            const int k0   = (c + 1) * KB;
            const int nbuf = buf ^ 1;
            cp16(gA + k0, &As[nbuf][ar][ac]);
#pragma unroll
            for (int i = 0; i < 4; ++i)
                cp16(gB + (size_t)i * 64 * H + k0, &Bs[nbuf][ar + i * 64][ac]);
        }
#pragma unroll
        for (int kk = 0; kk < KB; kk += 4) {
            v2f a0, a1;
            a0.x = As[buf][m0][kk + kb];
            a0.y = As[buf][m0][kk + kb + 1];
            a1.x = As[buf][m0 + 16][kk + kb];
            a1.y = As[buf][m0 + 16][kk + kb + 1];
            v2f b[4];
#pragma unroll
            for (int j = 0; j < 4; ++j) {
                b[j].x = Bs[buf][nb + j * 16][kk + kb];
                b[j].y = Bs[buf][nb + j * 16][kk + kb + 1];
            }
#pragma unroll
            for (int j = 0; j < 4; ++j) {
                acc[0][j] = __builtin_amdgcn_wmma_f32_16x16x4_f32(
                                false, a0, false, b[j], (short)0, acc[0][j], false, false);
                acc[1][j] = __builtin_amdgcn_wmma_f32_16x16x4_f32(
                                false, a1, false, b[j], (short)0, acc[1][j], false, false);
            }
        }
    }

    // epilogue: bias + relu; non-temporal store (512MB stream, not re-read
    // until kernel 3 -> keep x/W_enc in L2). C layout: lane L holds N=L&15,
    // M = vgpr + 8*(L>>4).
    const int rbase = rowA + mw * 32 + 8 * (lane >> 4);
#pragma unroll
    for (int mi = 0; mi < 2; ++mi) {
#pragma unroll
        for (int j = 0; j < 4; ++j) {
            const int col = colB + nw * 64 + j * 16 + (lane & 15);
            const float be = b_enc[col];
#pragma unroll
            for (int r = 0; r < 8; ++r) {
                const int row = rbase + mi * 16 + r;
                float v = acc[mi][j][r] + be;
                v = v > 0.f ? v : 0.f;
                __builtin_nontemporal_store(v, latent + (size_t)row * DSAE + col);
            }
        }
    }
}

// ---------------------------------------------------------------------------
// Kernel 3: per-row exact top-163 via 4-round radix select on float bits
// (valid: latents >= 0 after relu), parallel suffix-scan over the 256-bin
// histogram, then sparse decode with W_dec^T. One WG (256 threads) per row.
// ---------------------------------------------------------------------------
__global__ __launch_bounds__(256) void sae_select_decode(
    const float* __restrict__ latent, const float* __restrict__ WdT,
    const float* __restrict__ b_dec, float* __restrict__ out)
{
    __shared__ unsigned hist[256];
    __shared__ unsigned sh_prefix;
    __shared__ int sh_need;
    __shared__ int sh_cntHi, sh_cntEq, sh_base;
    __shared__ int   sidx[KTOP];
    __shared__ float sval[KTOP];

    const int row = blockIdx.x;
    const int t   = threadIdx.x;
    const float* lrow = latent + (size_t)row * DSAE;

    if (t == 0) { sh_prefix = 0u; sh_need = KTOP; }
    __syncthreads();

    // ---- radix select: find bit pattern of the k-th largest value ----
    for (int p = 24; p >= 0; p -= 8) {
        hist[t] = 0u;
        __syncthreads();
        const unsigned prefix = sh_prefix;
        const int      need   = sh_need;
#pragma unroll 4
        for (int i = 0; i < DSAE / 256; ++i) {
            const unsigned key = __float_as_uint(lrow[t + i * 256]);
            const bool in = (p == 24) || ((key >> (p + 8)) == (prefix >> (p + 8)));
            if (in) atomicAdd(&hist[(key >> p) & 255u], 1u);
        }
        __syncthreads();
        // parallel suffix scan: hist[t] <- sum of bins >= t
        for (int off = 1; off < 256; off <<= 1) {
            const unsigned v = (t + off < 256) ? hist[t + off] : 0u;
            __syncthreads();
            hist[t] += v;
            __syncthreads();
        }
        const int s     = (int)hist[t];
        const int snext = (t < 255) ? (int)hist[t + 1] : 0;
        if (s >= need && snext < need) {      // exactly one thread
            sh_prefix = prefix | ((unsigned)t << p);
            sh_need   = need - snext;         // still needed among ties in this byte
        }
        __syncthreads();
    }
    const unsigned T  = sh_prefix;  // bits of k-th largest value
    const int     rEq = sh_need;    // how many ==T to keep

    if (t == 0) { sh_cntHi = 0; sh_cntEq = 0; }
    __syncthreads();

    // ---- compact strictly-greater values ----
#pragma unroll 4
    for (int i = 0; i < DSAE / 256; ++i) {
        const int e = t + i * 256;
        const float v = lrow[e];
        if (__float_as_uint(v) > T) {
            const int pos = atomicAdd(&sh_cntHi, 1);
            if (pos < KTOP) { sidx[pos] = e; sval[pos] = v; }
        }
    }
    __syncthreads();
    if (t == 0) sh_base = sh_cntHi;
    __syncthreads();
    const int base = sh_base;

    // ---- append ties (==T) up to rEq ----
#pragma unroll 4
    for (int i = 0; i < DSAE / 256; ++i) {
        const int e = t + i * 256;
        const float v = lrow[e];
        if (__float_as_uint(v) == T) {
            const int pos  = atomicAdd(&sh_cntEq, 1);
            const int slot = base + pos;
            if (pos < rEq && slot < KTOP) { sidx[slot] = e; sval[slot] = v; }
        }
    }
    __syncthreads();
    int total = base + (sh_cntEq < rEq ? sh_cntEq : rEq);
    if (total > KTOP) total = KTOP;

    // ---- sparse decode: out[row,h] = b_dec[h] + sum_a val_a * WdT[j_a, h] ----
    const int h0 = t * 8;
    float acc[8];
    {
        const float4 b0 = *(const float4*)(b_dec + h0);
        const float4 b1 = *(const float4*)(b_dec + h0 + 4);
        acc[0] = b0.x; acc[1] = b0.y; acc[2] = b0.z; acc[3] = b0.w;
        acc[4] = b1.x; acc[5] = b1.y; acc[6] = b1.z; acc[7] = b1.w;
    }
    for (int a = 0; a < total; ++a) {
        const float v = sval[a];
        const float4* wd = (const float4*)(WdT + (size_t)sidx[a] * H + h0);
        const float4 w0 = wd[0];
        const float4 w1 = wd[1];
        acc[0] += v * w0.x; acc[1] += v * w0.y; acc[2] += v * w0.z; acc[3] += v * w0.w;
        acc[4] += v * w1.x; acc[5] += v * w1.y; acc[6] += v * w1.z; acc[7] += v * w1.w;
    }
    float4* o = (float4*)(out + (size_t)row * H + h0);
    o[0] = make_float4(acc[0], acc[1], acc[2], acc[3]);
    o[1] = make_float4(acc[4], acc[5], acc[6], acc[7]);
}

// ---------------------------------------------------------------------------
extern "C" void kernel_launch(void* const* d_in, const int* in_sizes, int n_in,
                              void* d_out, int out_size, void* d_ws, size_t ws_size,
                              hipStream_t stream)
{
    const float* x     = (const float*)d_in[0];
    const float* W_enc = (const float*)d_in[1];
    const float* b_enc = (const float*)d_in[2];
    const float* W_dec = (const float*)d_in[3];
    const float* b_dec = (const float*)d_in[4];
    float* out = (float*)d_out;

    float* latent = (float*)d_ws;                         // 8192*16384*4 = 512MB
    float* WdT    = latent + (size_t)NROWS * DSAE;        // 16384*2048*4 = 128MB

    sae_transpose_wdec<<<dim3(DSAE / 32, H / 32), 256, 0, stream>>>(W_dec, WdT);
    sae_encode<<<dim3(DSAE / BN, NROWS / BM), 256, 0, stream>>>(x, W_enc, b_enc, latent);
    sae_select_decode<<<NROWS, 256, 0, stream>>>(latent, WdT, b_dec, out);
}